// FAGCN_29291676959276
// MI455X (gfx1250) — compile-verified
//
#include <hip/hip_runtime.h>
#include <hip/hip_bf16.h>

typedef __attribute__((ext_vector_type(2))) float v2f;
typedef __attribute__((ext_vector_type(8))) float v8f;

#define D_DIM 64

// ---------------------------------------------------------------------------
// Kernel 1: per-node scalar scores a_src = x@w_src, a_dst = x@w_dst
// One wave32 per 16-node tile. fp32 WMMA (16x16x4) -> exact fp32 math.
// B matrix: column 0 = w_src, column 1 = w_dst, other 14 columns zero.
// B fragments are built branch-free (ds_load + cndmask) before the WMMA loop
// so the hot loop is pure ds_load_b64 (A) + v_wmma.
// ---------------------------------------------------------------------------
__global__ __launch_bounds__(32) void fagcn_node_scores(
    const float* __restrict__ x,
    const float* __restrict__ w_src,
    const float* __restrict__ w_dst,
    float* __restrict__ a_src,
    float* __restrict__ a_dst,
    int n)
{
    // 16 rows x 64 cols, padded to 68 floats/row -> conflict-free WMMA fetch
    __shared__ float tile[16][68];
    __shared__ float wbuf[2][64];

    const int lane = threadIdx.x;          // 0..31
    const int base = blockIdx.x * 16;      // first node of this tile

    // Stage weight vectors into LDS (coalesced, once per block).
    wbuf[0][lane]      = w_src[lane];
    wbuf[0][lane + 32] = w_src[lane + 32];
    wbuf[1][lane]      = w_dst[lane];
    wbuf[1][lane + 32] = w_dst[lane + 32];

    // Coalesced load of the 16x64 fp32 tile into LDS (float4 per lane-iter).
    #pragma unroll
    for (int it = 0; it < 8; ++it) {
        int idx = it * 32 + lane;          // 0..255
        int r   = idx >> 4;                // 0..15
        int c4  = (idx & 15) << 2;         // 0,4,...,60
        int row = base + r;
        float4 v = make_float4(0.f, 0.f, 0.f, 0.f);
        if (row < n) v = *(const float4*)(x + (size_t)row * D_DIM + c4);
        *(float4*)&tile[r][c4] = v;
    }
    __syncthreads();

    const int M    = lane & 15;            // row within tile (A) / col N (B)
    const int koff = (lane >> 4) << 1;     // 0 for lanes 0-15, 2 for 16-31

    // Build all 16 B fragments branch-free: clamp row select, then mask to 0.
    const bool isw = (M < 2);
    const int  wm  = isw ? M : 0;          // safe LDS row for every lane
    v2f bfrag[16];
    #pragma unroll
    for (int kk = 0; kk < 16; ++kk) {
        const int k0 = 4 * kk + koff;
        float b0 = wbuf[wm][k0];
        float b1 = wbuf[wm][k0 + 1];
        bfrag[kk].x = isw ? b0 : 0.f;
        bfrag[kk].y = isw ? b1 : 0.f;
    }

    v8f c = {};
    #pragma unroll
    for (int kk = 0; kk < 16; ++kk) {
        const int k0 = 4 * kk + koff;
        // A: 16x4 slice of x tile (lane<16 -> K0/K1, lane>=16 -> K2/K3)
        v2f a;
        a.x = tile[M][k0];
        a.y = tile[M][k0 + 1];
        // D = A*B + C   (v_wmma_f32_16x16x4_f32)
        c = __builtin_amdgcn_wmma_f32_16x16x4_f32(
                false, a, false, bfrag[kk], (short)0, c, false, false);
    }

    // D layout: VGPR v -> (M=v, N=lane) lanes 0-15 ; (M=v+8, N=lane-16) 16-31.
    // Column N=0 => a_src, column N=1 => a_dst.
    const int nsel  = lane & 15;
    const int rbase = base + ((lane >> 4) ? 8 : 0);
    if (nsel < 2) {
        float* outp = (nsel == 0) ? a_src : a_dst;
        #pragma unroll
        for (int v = 0; v < 8; ++v) {
            int row = rbase + v;
            if (row < n) outp[row] = c[v];
        }
    }
}

// ---------------------------------------------------------------------------
// Kernel 2: zero the output accumulator (harness poisons d_out).
// ---------------------------------------------------------------------------
__global__ __launch_bounds__(256) void fagcn_zero(float4* __restrict__ p, int n4)
{
    int i = blockIdx.x * blockDim.x + threadIdx.x;
    if (i < n4) p[i] = make_float4(0.f, 0.f, 0.f, 0.f);
}

// ---------------------------------------------------------------------------
// Kernel 3: fused edge attention + scatter.
// One wave32 per edge: wave-uniform index loads (scalarized), lane j owns a
// float2 of the 64-wide row -> fully coalesced 256B gather + coalesced f32
// atomics that resolve in the 192MB L2 (h and x are both L2-resident).
// ---------------------------------------------------------------------------
__global__ __launch_bounds__(256) void fagcn_edge_scatter(
    const float* __restrict__ x,
    const float* __restrict__ ew,
    const int* __restrict__ src_idx,
    const int* __restrict__ dst_idx,
    const float* __restrict__ a_src,
    const float* __restrict__ a_dst,
    float* __restrict__ h,
    int num_edges)
{
    const int wid  = (blockIdx.x * blockDim.x + threadIdx.x) >> 5;  // edge id
    const int lane = threadIdx.x & 31;
    if (wid >= num_edges) return;

    const int s = src_idx[wid];     // wave-uniform
    const int d = dst_idx[wid];     // wave-uniform

    const float a = tanhf(a_src[s] + a_dst[d]) * ew[wid];

    const float2 xv = *(const float2*)(x + (size_t)s * D_DIM + 2 * lane);
    float* hp = h + (size_t)d * D_DIM + 2 * lane;
    atomicAdd(hp + 0, a * xv.x);
    atomicAdd(hp + 1, a * xv.y);
}

// ---------------------------------------------------------------------------
// Launch: inputs = { x[N*64], w_src[64], w_dst[64], ew[E], src_idx[E], dst_idx[E] }
// ---------------------------------------------------------------------------
extern "C" void kernel_launch(void* const* d_in, const int* in_sizes, int n_in,
                              void* d_out, int out_size, void* d_ws, size_t ws_size,
                              hipStream_t stream)
{
    const float* x     = (const float*)d_in[0];
    const float* w_src = (const float*)d_in[1];
    const float* w_dst = (const float*)d_in[2];
    const float* ew    = (const float*)d_in[3];
    const int*   s_idx = (const int*)d_in[4];
    const int*   d_idx = (const int*)d_in[5];
    float*       h     = (float*)d_out;

    const int N = in_sizes[0] / D_DIM;   // 100000
    const int E = in_sizes[3];           // 1280000

    float* a_src = (float*)d_ws;
    float* a_dst = a_src + N;

    // Zero the output accumulator (scatter-add target).
    {
        int n4 = (N * D_DIM) / 4;
        int blocks = (n4 + 255) / 256;
        fagcn_zero<<<blocks, 256, 0, stream>>>((float4*)h, n4);
    }

    // Node scores via fp32 WMMA: one wave per 16 nodes.
    {
        int blocks = (N + 15) / 16;
        fagcn_node_scores<<<blocks, 32, 0, stream>>>(x, w_src, w_dst,
                                                     a_src, a_dst, N);
    }

    // Fused edge kernel: one wave per edge.
    {
        long long threads = (long long)E * 32;
        int blocks = (int)((threads + 255) / 256);
        fagcn_edge_scatter<<<blocks, 256, 0, stream>>>(x, ew, s_idx, d_idx,
                                                       a_src, a_dst, h, E);
    }
}